// Blurring_55224689492344
// MI455X (gfx1250) — compile-verified
//
#include <hip/hip_runtime.h>
#include <stdint.h>

// ---------------------------------------------------------------------------
// 3x3 median blur, (64,224,224,3) f32 NHWC, replicated borders.
// Data path: Tensor Data Mover (tensor_load_to_lds) stages an 18-row strip
// into LDS once; 8 wave32 compute medians via a min/max network.
// ---------------------------------------------------------------------------

typedef __attribute__((ext_vector_type(4))) unsigned int u32x4;
typedef __attribute__((ext_vector_type(8))) int          i32x8;
typedef __attribute__((ext_vector_type(4))) int          i32x4;

#define BDIM  256
#define BATCH 64
#define HH    224
#define WW    224
#define CC    3
#define TH    16                 // output rows per workgroup (224 % 16 == 0)
#define ROWF  (WW * CC)          // 672 floats per row
#define ROWB  (ROWF * 4)         // 2688 bytes per row
#define NROWS (TH + 2)           // 18 input rows staged (with halo)

// --- TDM: load `nrows` consecutive rows (stride ROWF dwords) into LDS -------
// D# layout per CDNA5 ISA ch.8 (groups 0/1; groups 2/3 zero for <=2D tiles).
__device__ __forceinline__ void tdm_load_rows(uint32_t lds_byte,
                                              const void* gptr, int nrows) {
    uint64_t ga = (uint64_t)(uintptr_t)gptr;
    u32x4 g0;
    g0[0] = 1u;                                   // count=1 (valid), no gather
    g0[1] = lds_byte;                             // lds_addr (bytes)
    g0[2] = (uint32_t)ga;                         // global_addr[31:0]
    g0[3] = (uint32_t)(ga >> 32) | (2u << 30);    // global_addr[56:32] | type=2

    uint32_t n = (uint32_t)nrows & 0xFFFFu;
    i32x8 g1;
    g1[0] = (int)(2u << 16);                      // wg_mask=0, data_size=4B
    g1[1] = (int)(((uint32_t)ROWF) << 16);        // tensor_dim0[15:0]
    g1[2] = (int)(n << 16);                       // dim0[31:16]=0 | tensor_dim1 lo
    g1[3] = (int)(((uint32_t)ROWF) << 16);        // dim1 hi=0 | tile_dim0=672
    g1[4] = (int)n;                               // tile_dim1=nrows, tile_dim2=0
    g1[5] = (int)(uint32_t)ROWF;                  // tensor_dim0_stride lo32
    g1[6] = 0;                                    // stride0 hi, stride1 lo
    g1[7] = 0;                                    // stride1 hi

    i32x4 z4 = {0, 0, 0, 0};                      // groups 2/3 unused (<=2D)
    i32x8 z8 = {0, 0, 0, 0, 0, 0, 0, 0};          // extra operand (6-arg form)
    __builtin_amdgcn_tensor_load_to_lds(g0, g1, z4, z4, z8, 0);
}

// --- median-of-9 via column sort + med3 combine (~30 v_min/v_max) -----------
__device__ __forceinline__ void s2(float& a, float& b) {
    float t = fminf(a, b);
    b = fmaxf(a, b);
    a = t;
}
__device__ __forceinline__ float med3(float a, float b, float c) {
    return fmaxf(fminf(a, b), fminf(fmaxf(a, b), c));
}
__device__ __forceinline__ float median9(float v0, float v1, float v2,
                                         float v3, float v4, float v5,
                                         float v6, float v7, float v8) {
    s2(v0, v1); s2(v1, v2); s2(v0, v1);           // sort triple 0
    s2(v3, v4); s2(v4, v5); s2(v3, v4);           // sort triple 1
    s2(v6, v7); s2(v7, v8); s2(v6, v7);           // sort triple 2
    float lo = fmaxf(fmaxf(v0, v3), v6);          // max of minima
    float hi = fminf(fminf(v2, v5), v8);          // min of maxima
    float md = med3(v1, v4, v7);                  // median of medians
    return med3(lo, md, hi);
}

__global__ __launch_bounds__(BDIM)
void median3x3_kernel(const float* __restrict__ x, float* __restrict__ out) {
    __shared__ float tile[NROWS * ROWF];          // 48384 B

    const int ht = blockIdx.x;
    const int b  = blockIdx.y;
    const int h0 = ht * TH;

    // Generic LDS pointers carry the aperture in the high 32 bits; the low
    // 32 bits are the group-relative LDS byte address (ISA 10.2).
    const uint32_t lds0 = (uint32_t)(uintptr_t)(&tile[0]);

    if (threadIdx.x == 0) {
        int rf = h0 - 1;  if (rf < 0) rf = 0;
        int rl = h0 + TH; if (rl > HH - 1) rl = HH - 1;
        int n  = rl - rf + 1;                     // 17 or 18 in-bounds rows
        int ldsrow = rf - (h0 - 1);               // 0 interior, 1 at top edge
        const char* base = (const char*)x + (size_t)(b * HH) * ROWB;

        tdm_load_rows(lds0 + (uint32_t)(ldsrow * ROWB),
                      base + (size_t)rf * ROWB, n);
        if (h0 == 0)                              // replicate top border row
            tdm_load_rows(lds0, base, 1);
        if (h0 + TH == HH)                        // replicate bottom border row
            tdm_load_rows(lds0 + (uint32_t)((TH + 1) * ROWB),
                          base + (size_t)(HH - 1) * ROWB, 1);

        __builtin_amdgcn_s_wait_tensorcnt(0);
    }
    __syncthreads();

    const size_t obase = (size_t)(b * HH + h0) * ROWF;
    for (int e = threadIdx.x; e < TH * ROWF; e += BDIM) {
        const int hr  = e / ROWF;
        const int rem = e - hr * ROWF;            // float index within row
        const int il  = (rem >= CC)          ? rem - CC : rem;  // left  (clamped)
        const int ir  = (rem < (ROWF - CC))  ? rem + CC : rem;  // right (clamped)

        const float* r0 = tile + hr * ROWF;
        const float* r1 = r0 + ROWF;
        const float* r2 = r1 + ROWF;

        out[obase + e] = median9(r0[il], r0[rem], r0[ir],
                                 r1[il], r1[rem], r1[ir],
                                 r2[il], r2[rem], r2[ir]);
    }
}

extern "C" void kernel_launch(void* const* d_in, const int* in_sizes, int n_in,
                              void* d_out, int out_size, void* d_ws, size_t ws_size,
                              hipStream_t stream) {
    const float* x   = (const float*)d_in[0];
    float*       out = (float*)d_out;
    dim3 grid(HH / TH, BATCH);                    // 14 x 64 = 896 workgroups
    median3x3_kernel<<<grid, BDIM, 0, stream>>>(x, out);
    (void)in_sizes; (void)n_in; (void)out_size; (void)d_ws; (void)ws_size;
}